// GridGNN_11897059409948
// MI455X (gfx1250) — compile-verified
//
#include <hip/hip_runtime.h>
#include <hip/hip_bf16.h>

typedef __attribute__((ext_vector_type(2))) float v2f;
typedef __attribute__((ext_vector_type(8))) float v8f;

#define F 64          // feature width (F_IN == F_HID == 64)
#define NGRAPH 64
#define NACT 3

// ---------------------------------------------------------------------------
// H[N x 64] = X[N x 64] @ W[64 x 64]   using V_WMMA_F32_16X16X4_F32 (fp32 MMA)
// One wave32 owns a 16-row stripe and produces all 64 output columns:
// 4 accumulators, 4 WMMAs per K-step (K=64 in steps of 4) -> 64 wmma/wave,
// A tile loaded once per K-step (amortized over 4 column tiles).
//
// A (16x4 f32): lanes 0-15 hold M=lane with {v0,v1}={K0,K1};
//               lanes 16-31 hold M=lane-16 with {v0,v1}={K2,K3}.
// B (4x16 f32): mirrored — lanes 0-15 hold N=lane with {v0,v1}={K0,K1};
//               lanes 16-31 hold N=lane-16 with {v0,v1}={K2,K3}.
// C/D (16x16 f32): vgpr v: lanes 0-15 -> M=v, lanes 16-31 -> M=v+8; N=lane&15.
// ---------------------------------------------------------------------------
__global__ __launch_bounds__(256) void gcn_gemm64_wmma(
    const float* __restrict__ X, const float* __restrict__ W,
    float* __restrict__ H, int N) {
  const int wave = (blockIdx.x * blockDim.x + threadIdx.x) >> 5;
  const int lane = threadIdx.x & 31;
  const int ntiles = (N + 15) >> 4;
  if (wave >= ntiles) return;
  const int m0 = wave << 4;
  const int lrow  = lane & 15;
  const int khalf = (lane >> 4) << 1;       // 0 (lanes 0-15) or 2 (16-31)

  int arow = m0 + lrow;                     // clamp loads; stores are guarded
  if (arow >= N) arow = N - 1;

  v8f acc0 = {}, acc1 = {}, acc2 = {}, acc3 = {};
#pragma unroll
  for (int k0 = 0; k0 < F; k0 += 4) {
    const int k = k0 + khalf;
    v2f a;
    a.x = X[(size_t)arow * F + k];
    a.y = X[(size_t)arow * F + k + 1];
    const float* w0 = W + (size_t)k * F + lrow;
    const float* w1 = W + (size_t)(k + 1) * F + lrow;
    v2f b0, b1, b2, b3;
    b0.x = w0[0];  b0.y = w1[0];
    b1.x = w0[16]; b1.y = w1[16];
    b2.x = w0[32]; b2.y = w1[32];
    b3.x = w0[48]; b3.y = w1[48];
    acc0 = __builtin_amdgcn_wmma_f32_16x16x4_f32(false, a, false, b0,
                                                 (short)0, acc0, false, false);
    acc1 = __builtin_amdgcn_wmma_f32_16x16x4_f32(false, a, false, b1,
                                                 (short)0, acc1, false, false);
    acc2 = __builtin_amdgcn_wmma_f32_16x16x4_f32(false, a, false, b2,
                                                 (short)0, acc2, false, false);
    acc3 = __builtin_amdgcn_wmma_f32_16x16x4_f32(false, a, false, b3,
                                                 (short)0, acc3, false, false);
  }

  const int rbase = m0 + ((lane >> 4) << 3);
#pragma unroll
  for (int v = 0; v < 8; ++v) {
    const int row = rbase + v;
    if (row < N) {
      float* hp = H + (size_t)row * F + lrow;
      hp[0]  = acc0[v];
      hp[16] = acc1[v];
      hp[32] = acc2[v];
      hp[48] = acc3[v];
    }
  }
}

// --------------------------- degree / norm ---------------------------------
__global__ void deg_init(float* __restrict__ deg, int N) {
  int i = blockIdx.x * blockDim.x + threadIdx.x;
  if (i < N) deg[i] = 1.0f;                 // self loop
}

__global__ void deg_accum(const int* __restrict__ dst, float* __restrict__ deg,
                          int E) {
  int e = blockIdx.x * blockDim.x + threadIdx.x;
  if (e < E) atomicAdd(&deg[dst[e]], 1.0f);
}

__global__ void deg_rsqrt(float* __restrict__ deg, int N) {
  int i = blockIdx.x * blockDim.x + threadIdx.x;
  if (i < N) deg[i] = rsqrtf(deg[i]);       // deg >= 1
}

// --------------------------- edge scatter ----------------------------------
// One wave32 per edge: the edge index is forced wave-uniform via
// readfirstlane so src/dst/dinv loads become scalar (SMEM) and the coef is
// SALU math; each lane gathers a float2 of h[src] and issues two coalesced
// global_atomic_add_f32 into agg[dst].
__global__ void edge_scatter(const float* __restrict__ h,
                             const float* __restrict__ dinv,
                             const int* __restrict__ src,
                             const int* __restrict__ dst,
                             float* __restrict__ agg, int E) {
  const long long wid =
      ((long long)blockIdx.x * blockDim.x + threadIdx.x) >> 5;
  const int lane = threadIdx.x & 31;
  if (wid >= E) return;
  const int e = __builtin_amdgcn_readfirstlane((int)wid);
  const int s = src[e];
  const int d = dst[e];
  const float c = dinv[s] * dinv[d];
  const float2 hv = *(const float2*)(h + (size_t)s * F + lane * 2);
  float* ap = agg + (size_t)d * F + lane * 2;
  atomicAdd(ap,     hv.x * c);
  atomicAdd(ap + 1, hv.y * c);
}

// out = agg + hpre * dinv^2 + bias   (optionally ReLU); safe to alias out
// with hpre or agg since it's purely elementwise at the same index.
__global__ void gcn_finalize(const float* __restrict__ hpre,
                             const float* __restrict__ agg,
                             const float* __restrict__ dinv,
                             const float* __restrict__ bias,
                             float* __restrict__ out, int N, int do_relu) {
  long long t = (long long)blockIdx.x * blockDim.x + threadIdx.x;
  if (t >= (long long)N * F) return;
  const int i = (int)(t >> 6);
  const int f = (int)(t & (F - 1));
  const float di = dinv[i];
  float v = agg[t] + hpre[t] * (di * di) + bias[f];
  if (do_relu) v = fmaxf(v, 0.0f);
  out[t] = v;
}

// --------------------------- pooling + head --------------------------------
// One wave per node; batch id forced wave-uniform -> scalar load.
__global__ void pool_accum(const float* __restrict__ h,
                           const int* __restrict__ batch,
                           float* __restrict__ sums, float* __restrict__ cnt,
                           int N) {
  const long long wid =
      ((long long)blockIdx.x * blockDim.x + threadIdx.x) >> 5;
  const int lane = threadIdx.x & 31;
  if (wid >= N) return;
  const int i = __builtin_amdgcn_readfirstlane((int)wid);
  const int g = batch[i];
  const float2 hv = *(const float2*)(h + (size_t)i * F + lane * 2);
  float* sp = sums + (size_t)g * F + lane * 2;
  atomicAdd(sp,     hv.x);
  atomicAdd(sp + 1, hv.y);
  if (lane == 0) atomicAdd(&cnt[g], 1.0f);
}

__global__ void head_softmax(const float* __restrict__ sums,
                             const float* __restrict__ cnt,
                             const float* __restrict__ Wl,
                             const float* __restrict__ bl,
                             float* __restrict__ out) {
  const int g = threadIdx.x;
  if (g >= NGRAPH) return;
  const float inv = 1.0f / fmaxf(cnt[g], 1.0f);
  float logit[NACT];
#pragma unroll
  for (int j = 0; j < NACT; ++j) logit[j] = bl[j];
  for (int f = 0; f < F; ++f) {
    const float m = sums[(size_t)g * F + f] * inv;
#pragma unroll
    for (int j = 0; j < NACT; ++j) logit[j] += m * Wl[f * NACT + j];
  }
  float mx = logit[0];
#pragma unroll
  for (int j = 1; j < NACT; ++j) mx = fmaxf(mx, logit[j]);
  float ex[NACT], s = 0.0f;
#pragma unroll
  for (int j = 0; j < NACT; ++j) { ex[j] = __expf(logit[j] - mx); s += ex[j]; }
  const float is = 1.0f / s;
#pragma unroll
  for (int j = 0; j < NACT; ++j) out[g * NACT + j] = ex[j] * is;
}

// ---------------------------------------------------------------------------
extern "C" void kernel_launch(void* const* d_in, const int* in_sizes, int n_in,
                              void* d_out, int out_size, void* d_ws,
                              size_t ws_size, hipStream_t stream) {
  const float* x     = (const float*)d_in[0];
  const int*   edge  = (const int*)d_in[1];   // [2, E] : src row then dst row
  const int*   batch = (const int*)d_in[2];
  const float* W1    = (const float*)d_in[3];
  const float* b1    = (const float*)d_in[4];
  const float* W2    = (const float*)d_in[5];
  const float* b2    = (const float*)d_in[6];
  const float* Wl    = (const float*)d_in[7];
  const float* bl    = (const float*)d_in[8];

  const int N = in_sizes[0] / F;              // 100000
  const int E = in_sizes[1] / 2;              // 1600000
  const int* src = edge;
  const int* dst = edge + E;

  // Workspace layout: two N*64 fp32 buffers (ping-pong), dinv, pool sums+cnt.
  float* bufA = (float*)d_ws;
  float* bufB = bufA + (size_t)N * F;
  float* dinv = bufB + (size_t)N * F;
  float* pool = dinv + N;
  float* cnt  = pool + (size_t)NGRAPH * F;

  const int nodeBlocks = (N + 255) / 256;
  const int edgeBlocks = (E + 255) / 256;
  const long long tN = (long long)N * F;
  const int tNBlocks = (int)((tN + 255) / 256);
  const int scatterBlocks = (int)(((long long)E * 32 + 255) / 256);
  const int poolBlocks = (int)(((long long)N * 32 + 255) / 256);
  const int gemmWaves = (N + 15) / 16;        // one wave per 16-row stripe
  const int gemmBlocks = (gemmWaves * 32 + 255) / 256;

  // Symmetric normalization coefficients.
  deg_init<<<nodeBlocks, 256, 0, stream>>>(dinv, N);
  deg_accum<<<edgeBlocks, 256, 0, stream>>>(dst, dinv, E);
  deg_rsqrt<<<nodeBlocks, 256, 0, stream>>>(dinv, N);

  // ---- Layer 1: h = relu(gcn(x, W1, b1)) ----
  gcn_gemm64_wmma<<<gemmBlocks, 256, 0, stream>>>(x, W1, bufA, N);
  hipMemsetAsync(bufB, 0, (size_t)N * F * sizeof(float), stream);
  edge_scatter<<<scatterBlocks, 256, 0, stream>>>(bufA, dinv, src, dst, bufB, E);
  gcn_finalize<<<tNBlocks, 256, 0, stream>>>(bufA, bufB, dinv, b1, bufA, N, 1);

  // ---- Layer 2: h = gcn(h, W2, b2) ----
  gcn_gemm64_wmma<<<gemmBlocks, 256, 0, stream>>>(bufA, W2, bufB, N);
  hipMemsetAsync(bufA, 0, (size_t)N * F * sizeof(float), stream);
  edge_scatter<<<scatterBlocks, 256, 0, stream>>>(bufB, dinv, src, dst, bufA, E);
  gcn_finalize<<<tNBlocks, 256, 0, stream>>>(bufB, bufA, dinv, b2, bufA, N, 0);

  // ---- Global mean pool + linear + softmax ----
  hipMemsetAsync(pool, 0, (size_t)(NGRAPH * F + NGRAPH) * sizeof(float), stream);
  pool_accum<<<poolBlocks, 256, 0, stream>>>(bufA, batch, pool, cnt, N);
  head_softmax<<<1, 64, 0, stream>>>(pool, cnt, Wl, bl, (float*)d_out);
}